// UCOEA_22548578304501
// MI455X (gfx1250) — compile-verified
//
#include <hip/hip_runtime.h>
#include <math.h>

#define BDIM 8
#define CH_IN 512
#define C2 256
#define HIN 128
#define HOUT 64
#define HW 4096          // 64*64
#define KCL 8
#define NGM 2
#define KM_ITERS 10
#define EM_ITERS 10
#define EPSF 1e-6f
#define LOG2PI 1.8378770664093453f

typedef __attribute__((ext_vector_type(2))) float v2f;
typedef __attribute__((ext_vector_type(8))) float v8f;
typedef __attribute__((ext_vector_type(4))) int   v4i;

#if defined(__has_builtin)
#if __has_builtin(__builtin_amdgcn_global_load_async_to_lds_b128) && \
    __has_builtin(__builtin_amdgcn_s_wait_asynccnt)
#define HAS_ASYNC_LDS 1
#endif
#endif
#ifndef HAS_ASYNC_LDS
#define HAS_ASYNC_LDS 0
#endif

// ---------------- Stage 1: grouped stride-2 3x3 conv (feature_group_count=256) ----
__global__ void conv_down_kernel(const float* __restrict__ sx,
                                 const float* __restrict__ wd,
                                 float* __restrict__ x) {
  int idx = blockIdx.x * blockDim.x + threadIdx.x;
  int ow = idx & 63;
  int oh = (idx >> 6) & 63;
  int co = (idx >> 12) & 255;
  int b  = idx >> 20;
  const float* w = wd + co * 18;        // (co, 2, 3, 3)
  float acc = 0.f;
#pragma unroll
  for (int i = 0; i < 2; ++i) {
    const float* sp = sx + ((size_t)b * CH_IN + 2 * co + i) * HIN * HIN;
#pragma unroll
    for (int kh = 0; kh < 3; ++kh) {
      int ih = 2 * oh + kh - 1;
      if (ih < 0 || ih >= HIN) continue;
#pragma unroll
      for (int kw = 0; kw < 3; ++kw) {
        int iw = 2 * ow + kw - 1;
        if (iw < 0 || iw >= HIN) continue;
        acc += sp[ih * HIN + iw] * w[(i * 3 + kh) * 3 + kw];
      }
    }
  }
  x[idx] = acc;
}

// ---------------- Stage 2: per-channel min/max normalize of batch 0 ----------------
__global__ void normalize_kernel(const float* __restrict__ x, float* __restrict__ xs) {
  int c = blockIdx.x, tid = threadIdx.x;
  const float* row = x + (size_t)c * HW;  // batch 0
  float mn = 1e30f, mx = -1e30f;
  for (int d = tid; d < HW; d += 256) { float v = row[d]; mn = fminf(mn, v); mx = fmaxf(mx, v); }
  __shared__ float smn[256], smx[256];
  smn[tid] = mn; smx[tid] = mx; __syncthreads();
  for (int s = 128; s > 0; s >>= 1) {
    if (tid < s) { smn[tid] = fminf(smn[tid], smn[tid + s]); smx[tid] = fmaxf(smx[tid], smx[tid + s]); }
    __syncthreads();
  }
  float lo = smn[0], range = fmaxf(smx[0] - smn[0], 1e-12f);
  for (int d = tid; d < HW; d += 256) xs[(size_t)c * HW + d] = (row[d] - lo) / range;
}

// ---------------- Stage 3: k-means (one block, thread i owns point i) --------------
__global__ void kmeans_kernel(const float* __restrict__ xs, const float* __restrict__ x0,
                              float* __restrict__ cent, float* __restrict__ Wmat,
                              int* __restrict__ slabel_g) {
  int tid = threadIdx.x;
  __shared__ int   labels[256];
  __shared__ float cnt[KCL], c2s[KCL], cs1[KCL], cs2[KCL], scnt[KCL];
  __shared__ int   invmap[KCL];
  const int init_idx[KCL] = {0, 36, 73, 109, 146, 182, 219, 255}; // round(linspace(0,255,8))
  for (int k = 0; k < KCL; ++k)
    for (int d = tid; d < HW; d += 256) cent[k * HW + d] = xs[(size_t)init_idx[k] * HW + d];
  __syncthreads();

  const float* myrow = xs + (size_t)tid * HW;
  float x2 = 0.f;
  for (int d = 0; d < HW; ++d) { float v = myrow[d]; x2 += v * v; }

  for (int iter = 0; iter <= KM_ITERS; ++iter) {
    if (tid < KCL) { float s = 0.f; const float* cr = cent + tid * HW;
      for (int d = 0; d < HW; ++d) s += cr[d] * cr[d]; c2s[tid] = s; }
    __syncthreads();
    float dot[KCL];
#pragma unroll
    for (int k = 0; k < KCL; ++k) dot[k] = 0.f;
    for (int d = 0; d < HW; ++d) {
      float v = myrow[d];
#pragma unroll
      for (int k = 0; k < KCL; ++k) dot[k] += v * cent[k * HW + d];
    }
    int best = 0; float bd = x2 - 2.f * dot[0] + c2s[0];
#pragma unroll
    for (int k = 1; k < KCL; ++k) { float dk = x2 - 2.f * dot[k] + c2s[k]; if (dk < bd) { bd = dk; best = k; } }
    labels[tid] = best;
    __syncthreads();
    if (iter == KM_ITERS) break;
    if (tid < KCL) { float c = 0.f; for (int i = 0; i < 256; ++i) c += (labels[i] == tid) ? 1.f : 0.f; cnt[tid] = c; }
    __syncthreads();
    for (int d = tid; d < HW; d += 256) {
      float acc[KCL];
#pragma unroll
      for (int k = 0; k < KCL; ++k) acc[k] = 0.f;
      for (int i = 0; i < 256; ++i) acc[labels[i]] += xs[(size_t)i * HW + d];
      for (int k = 0; k < KCL; ++k) if (cnt[k] > 0.f) cent[k * HW + d] = acc[k] / cnt[k];
    }
    __syncthreads();
  }

  // cluster means over raw x0, std over dims, sort clusters by std
  if (tid < KCL) {
    float c = 0.f; for (int i = 0; i < 256; ++i) c += (labels[i] == tid) ? 1.f : 0.f;
    cnt[tid] = c; cs1[tid] = 0.f; cs2[tid] = 0.f;
  }
  __syncthreads();
  for (int d = tid; d < HW; d += 256) {
    float acc[KCL];
#pragma unroll
    for (int k = 0; k < KCL; ++k) acc[k] = 0.f;
    for (int i = 0; i < 256; ++i) acc[labels[i]] += x0[(size_t)i * HW + d];
    for (int k = 0; k < KCL; ++k) {
      float m = acc[k] / fmaxf(cnt[k], 1.f);
      atomicAdd(&cs1[k], m); atomicAdd(&cs2[k], m * m);
    }
  }
  __syncthreads();
  if (tid == 0) {
    float v22[KCL];
    for (int k = 0; k < KCL; ++k) { float m = cs1[k] / (float)HW; v22[k] = sqrtf(fmaxf(cs2[k] / (float)HW - m * m, 0.f)); }
    int order[KCL]; bool used[KCL];
    for (int k = 0; k < KCL; ++k) used[k] = false;
    for (int r = 0; r < KCL; ++r) {
      int bsel = -1;
      for (int k = 0; k < KCL; ++k) if (!used[k] && (bsel < 0 || v22[k] < v22[bsel])) bsel = k;
      order[r] = bsel; used[bsel] = true;
    }
    for (int r = 0; r < KCL; ++r) invmap[order[r]] = r;
    for (int r = 0; r < KCL; ++r) scnt[r] = cnt[order[r]];
  }
  __syncthreads();
  int sl = invmap[labels[tid]];
  slabel_g[tid] = sl;
  for (int k = 0; k < KCL; ++k) Wmat[tid * KCL + k] = (sl == k) ? 1.f / fmaxf(scnt[k], 1.f) : 0.f;
}

// ---------------- Stage 4: einsum bchw,ck->bkhw as f32 WMMA GEMM -------------------
// One wave per 16x16 tile: D[k=0..15][p0..p0+15] = sum_c A[k][c] * X[c][p].
// A is zero-padded (rows 8..15) and pair-interleaved in LDS -> 1 ds_load_b64/step.
// The 256x16 X panel is staged to LDS via GLOBAL_LOAD_ASYNC_TO_LDS_B128 (ASYNCcnt,
// drained with s_wait_asynccnt 0), then consumed with unconditional ds loads; EXEC
// stays all-ones across every v_wmma_f32_16x16x4_f32 as the ISA requires.
__global__ void __launch_bounds__(32) v5_wmma_kernel(const float* __restrict__ x,
                                                     const float* __restrict__ Wmat,
                                                     float* __restrict__ wa) {
  __shared__ float Apack[(C2 / 2) * 32];   // 16 KB: [(c/2)][n*2+j] = W^T[n][c+j], 0-padded
  __shared__ float Xlds[C2 * 16];          // 16 KB: [c][n] panel, row-major
  int lane = threadIdx.x;
  int half = lane >> 4;     // lanes 16-31 carry K+2,K+3 slices (ISA 16x4 A layout)
  int n    = lane & 15;
  int p0 = blockIdx.x * 16;
  int b  = blockIdx.y;
  const float* xb = x + (size_t)b * C2 * HW;

  // Build zero-padded interleaved A fragments in LDS.
  for (int idx = lane; idx < (C2 / 2) * 32; idx += 32) {
    int c2i = idx >> 5, within = idx & 31;
    int nn = within >> 1, jj = within & 1;
    Apack[idx] = (nn < KCL) ? Wmat[(2 * c2i + jj) * KCL + nn] : 0.f;
  }

  // Stage X panel (256 channels x 16 pixels) into LDS.
  int r = lane >> 2, q = lane & 3;         // 8 rows x 4 quads per pass
#if HAS_ASYNC_LDS
  for (int cb = 0; cb < C2; cb += 8) {
    float* g = (float*)(xb + (size_t)(cb + r) * HW + p0 + 4 * q);
    float* l = &Xlds[(cb + r) * 16 + 4 * q];
    __builtin_amdgcn_global_load_async_to_lds_b128(
        (__attribute__((address_space(1))) v4i*)g,
        (__attribute__((address_space(3))) v4i*)l, 0, 0);
  }
  __builtin_amdgcn_s_wait_asynccnt(0);
  __syncthreads();
#else
  for (int cb = 0; cb < C2; cb += 8) {
    const float4 v = *(const float4*)(xb + (size_t)(cb + r) * HW + p0 + 4 * q);
    *(float4*)&Xlds[(cb + r) * 16 + 4 * q] = v;
  }
  __syncthreads();
#endif

  v8f acc = {0.f, 0.f, 0.f, 0.f, 0.f, 0.f, 0.f, 0.f};
  for (int c0 = 0; c0 < C2; c0 += 4) {
    int ka = c0 + 2 * half;
    v2f a = *(const v2f*)&Apack[(c0 / 2 + half) * 32 + n * 2];  // ds_load_b64
    v2f bb;
    bb.x = Xlds[(ka + 0) * 16 + n];                             // B[k][n] = X[ka][p0+n]
    bb.y = Xlds[(ka + 1) * 16 + n];
    acc = __builtin_amdgcn_wmma_f32_16x16x4_f32(false, a, false, bb,
                                                (short)0, acc, false, false);
  }
  if (lane < 16) {           // rows 0..7 valid -> wa[b][p][k] contiguous in k
    float* dst = wa + ((size_t)b * HW + p0 + lane) * KCL;
#pragma unroll
    for (int rr = 0; rr < KCL; ++rr) dst[rr] = acc[rr];
  }
}

// ---------------- Stage 5: GMM EM (N=2) + gate per batch ---------------------------
__global__ void gmm_gate_kernel(const float* __restrict__ wa, const float* __restrict__ w_conv,
                                const int* __restrict__ slabel, float* __restrict__ scaleArr) {
  int b = blockIdx.x, tid = threadIdx.x;
  const float* db = wa + (size_t)b * HW * KCL;
  __shared__ float mu[NGM][KCL], var[NGM][KCL], logw[NGM], logdet[NGM];
  __shared__ float Nk[NGM], Sm[NGM][KCL], Se[NGM][KCL];
  __shared__ float s1[KCL], s2[KCL], ys[KCL];
  if (tid < KCL) { s1[tid] = 0.f; s2[tid] = 0.f; }
  __syncthreads();
  {
    float ls1[KCL], ls2[KCL];
#pragma unroll
    for (int k = 0; k < KCL; ++k) { ls1[k] = 0.f; ls2[k] = 0.f; }
    for (int j = 0; j < 16; ++j) {
      const float* dp = db + (size_t)(tid + j * 256) * KCL;
#pragma unroll
      for (int k = 0; k < KCL; ++k) { float v = dp[k]; ls1[k] += v; ls2[k] += v * v; }
    }
    for (int k = 0; k < KCL; ++k) { atomicAdd(&s1[k], ls1[k]); atomicAdd(&s2[k], ls2[k]); }
  }
  __syncthreads();
  if (tid == 0) {
    float ld = 0.f;
    for (int k = 0; k < KCL; ++k) {
      float m = s1[k] / (float)HW;
      float v = s2[k] / (float)HW - m * m + EPSF;
      var[0][k] = v; var[1][k] = v;
      mu[0][k] = db[k];                                 // pick = [0, 4095]
      mu[1][k] = db[(size_t)(HW - 1) * KCL + k];
      ld += logf(v);
    }
    logdet[0] = ld; logdet[1] = ld;
    logw[0] = logf(0.5f); logw[1] = logf(0.5f);
  }
  __syncthreads();
  for (int it = 0; it < EM_ITERS; ++it) {
    if (tid < NGM) Nk[tid] = 0.f;
    if (tid < NGM * KCL) { ((float*)Sm)[tid] = 0.f; ((float*)Se)[tid] = 0.f; }
    __syncthreads();
    float rN[NGM], rSm[NGM][KCL], rSe[NGM][KCL];
#pragma unroll
    for (int nn = 0; nn < NGM; ++nn) { rN[nn] = 0.f;
      for (int k = 0; k < KCL; ++k) { rSm[nn][k] = 0.f; rSe[nn][k] = 0.f; } }
    for (int j = 0; j < 16; ++j) {
      const float* dp = db + (size_t)(tid + j * 256) * KCL;
      float xv[KCL];
#pragma unroll
      for (int k = 0; k < KCL; ++k) xv[k] = dp[k];
      float lp[NGM];
#pragma unroll
      for (int nn = 0; nn < NGM; ++nn) {
        float d2 = 0.f;
        for (int k = 0; k < KCL; ++k) { float df = xv[k] - mu[nn][k]; d2 += df * df / var[nn][k]; }
        lp[nn] = -0.5f * (d2 + logdet[nn] + (float)KCL * LOG2PI) + logw[nn];
      }
      float m = fmaxf(lp[0], lp[1]);
      float e0 = expf(lp[0] - m), e1 = expf(lp[1] - m);
      float inv = 1.f / (e0 + e1);
      float r[NGM] = {e0 * inv, e1 * inv};
#pragma unroll
      for (int nn = 0; nn < NGM; ++nn) {
        rN[nn] += r[nn];
        for (int k = 0; k < KCL; ++k) { float v = r[nn] * xv[k]; rSm[nn][k] += v; rSe[nn][k] += v * xv[k]; }
      }
    }
    for (int nn = 0; nn < NGM; ++nn) {
      atomicAdd(&Nk[nn], rN[nn]);
      for (int k = 0; k < KCL; ++k) { atomicAdd(&Sm[nn][k], rSm[nn][k]); atomicAdd(&Se[nn][k], rSe[nn][k]); }
    }
    __syncthreads();
    if (tid == 0) {
      for (int nn = 0; nn < NGM; ++nn) {
        float ld = 0.f;
        for (int k = 0; k < KCL; ++k) {
          float m2 = Sm[nn][k] / Nk[nn];
          float e2 = Se[nn][k] / Nk[nn];
          float v = fmaxf(e2 - m2 * m2 + EPSF, EPSF);
          mu[nn][k] = m2; var[nn][k] = v; ld += logf(v);
        }
        logdet[nn] = ld;
        logw[nn] = logf(Nk[nn] / (float)HW);
      }
    }
    __syncthreads();
  }
  if (tid == 0) {
    for (int k = 0; k < KCL; ++k) {
      float b0, b1;                        // sort 2 comps by variance (stable asc)
      if (var[0][k] <= var[1][k]) { b0 = mu[0][k]; b1 = mu[1][k]; }
      else                        { b0 = mu[1][k]; b1 = mu[0][k]; }
      float z = b0 * w_conv[k * NGM + 0] + b1 * w_conv[k * NGM + 1];
      ys[k] = 1.f / (1.f + expf(-z));
    }
  }
  __syncthreads();
  scaleArr[b * C2 + tid] = 1.f + ys[slabel[tid]];
}

// ---------------- Stage 6: fused gate*x, channel-repeat*w_up, bilinear x2, +sx -----
__global__ void upsample_out_kernel(const float* __restrict__ x, const float* __restrict__ sx,
                                    const float* __restrict__ scaleArr, const float* __restrict__ w_up,
                                    float* __restrict__ out) {
  size_t idx = (size_t)blockIdx.x * blockDim.x + threadIdx.x;
  int ox = (int)(idx & 127);
  int oy = (int)((idx >> 7) & 127);
  int j  = (int)((idx >> 14) & 511);
  int b  = (int)(idx >> 23);
  int c  = j >> 1;
  float sy = oy * (63.f / 127.f);          // linspace(0, 63, 128)
  float sxf = ox * (63.f / 127.f);
  int y0 = (int)sy;  int y1 = min(y0 + 1, 63);  float fy = sy - y0;
  int x0 = (int)sxf; int x1 = min(x0 + 1, 63);  float fx = sxf - x0;
  const float* xp = x + ((size_t)b * C2 + c) * HOUT * HOUT;
  float v00 = xp[y0 * HOUT + x0], v01 = xp[y0 * HOUT + x1];
  float v10 = xp[y1 * HOUT + x0], v11 = xp[y1 * HOUT + x1];
  float val = (v00 * (1.f - fx) + v01 * fx) * (1.f - fy) + (v10 * (1.f - fx) + v11 * fx) * fy;
  float sc = scaleArr[b * C2 + c] * w_up[j];
  out[idx] = val * sc + sx[idx];
}

extern "C" void kernel_launch(void* const* d_in, const int* in_sizes, int n_in,
                              void* d_out, int out_size, void* d_ws, size_t ws_size,
                              hipStream_t stream) {
  const float* sx     = (const float*)d_in[0];
  const float* w_down = (const float*)d_in[1];
  const float* w_conv = (const float*)d_in[2];
  const float* w_up   = (const float*)d_in[3];
  float* out = (float*)d_out;

  float* ws       = (float*)d_ws;
  float* x        = ws;                               // 8*256*4096   =  8,388,608 f
  float* xs       = x + (size_t)BDIM * C2 * HW;       // 256*4096     =  1,048,576 f
  float* cent     = xs + (size_t)C2 * HW;             // 8*4096       =     32,768 f
  float* wa       = cent + KCL * HW;                  // 8*4096*8     =    262,144 f
  float* Wmat     = wa + (size_t)BDIM * HW * KCL;     // 256*8        =      2,048 f
  float* scaleArr = Wmat + C2 * KCL;                  // 8*256        =      2,048 f
  int*   slabel   = (int*)(scaleArr + BDIM * C2);     // 256 ints

  conv_down_kernel<<<(BDIM * C2 * HOUT * HOUT) / 256, 256, 0, stream>>>(sx, w_down, x);
  normalize_kernel<<<C2, 256, 0, stream>>>(x, xs);
  kmeans_kernel<<<1, 256, 0, stream>>>(xs, x, cent, Wmat, slabel);
  v5_wmma_kernel<<<dim3(HW / 16, BDIM), 32, 0, stream>>>(x, Wmat, wa);
  gmm_gate_kernel<<<BDIM, 256, 0, stream>>>(wa, w_conv, slabel, scaleArr);
  upsample_out_kernel<<<(BDIM * CH_IN * HIN * HIN) / 256, 256, 0, stream>>>(x, sx, scaleArr, w_up, out);
}